// GraphPooling_42099269435629
// MI455X (gfx1250) — compile-verified
//
#include <hip/hip_runtime.h>

// ---------------------------------------------------------------------------
// GraphPooling on MI455X (gfx1250):
//   pass1: scores = dot(mean_f x, W) + b          (reads 256MB, WMMA f32 dot)
//   csr  : invert segment_ids -> sorted lists     (deterministic, 1 block)
//   pass2: segmented softmax weights              (4 blocks, LDS atomics)
//   pass3: out[b,c] = sum_n w*x[b,n] gather-reduce (reads 256MB, writes 16MB)
// Memory-bound: ~528MB @ 23.3TB/s ~= 23us floor.
// ---------------------------------------------------------------------------

typedef float v2f __attribute__((ext_vector_type(2)));
typedef float v8f __attribute__((ext_vector_type(8)));

#define B_    4
#define NF_   4096
#define FM_   32
#define H_    128
#define NC_   256
#define TILE_ (FM_*H_)        // 4096 floats per fine node
#define NPB_  16              // nodes per block in pass 1
#define XS_   132             // padded LDS row stride (floats): 132%64=4 -> no bank conflicts

// ------------------------------ Pass 1 -------------------------------------
// scores[b,n] = (1/Fm) * sum_f dot(x[b,n,f,:], W) + bias
// Phase A: 256 threads stream-sum the 16-node tile over Fm into LDS (float4).
// Phase B: wave 0 runs 32 x V_WMMA_F32_16X16X4_F32 accumulations:
//   A[m][k] = W[k]        (all rows identical; documented 16x4 f32 A layout)
//   B[k][n] = xsum[n][k]  (4x16 layout mirrored from A: lanes 0-15 rows {K0,K1},
//                          lanes 16-31 rows {K2,K3}, col = lane%16)
//   => D[m][n] = score[n] in every row; lane n extracts D[0].
__global__ __launch_bounds__(256)
void k_scores(const float* __restrict__ x, const float* __restrict__ W,
              const float* __restrict__ bias, float* __restrict__ scores)
{
    __shared__ float xsum[NPB_ * XS_];
    __shared__ float Wl[H_];
    const int t  = threadIdx.x;
    const int n0 = blockIdx.x * NPB_;
    const int bb = blockIdx.y;

    if (t < H_) Wl[t] = W[t];

    // Phase A: 512 float4 positions, 2 per thread
    #pragma unroll
    for (int rep = 0; rep < 2; ++rep) {
        const int p  = t + rep * 256;
        const int n  = p >> 5;       // node within block tile (0..15)
        const int h4 = p & 31;       // float4 column (0..31)
        const float4* src = (const float4*)(x + (size_t)(bb * NF_ + n0 + n) * TILE_) + h4;
        float4 acc = {0.f, 0.f, 0.f, 0.f};
        #pragma unroll 8
        for (int f = 0; f < FM_; ++f) {
            float4 v = src[f * (H_ / 4)];
            acc.x += v.x; acc.y += v.y; acc.z += v.z; acc.w += v.w;
        }
        *(float4*)&xsum[n * XS_ + h4 * 4] = acc;
    }
    __syncthreads();

    if (t < 32) {                    // wave 0, EXEC all-ones (wave32)
        const int lane = t;
        const int half = lane >> 4;  // 0 -> K{0,1}, 1 -> K{2,3}
        const int n    = lane & 15;
        v8f c = {};
        #pragma unroll
        for (int k0 = 0; k0 < H_; k0 += 4) {
            v2f a, bm;
            a.x  = Wl[k0 + 2 * half + 0];
            a.y  = Wl[k0 + 2 * half + 1];
            bm.x = xsum[n * XS_ + k0 + 2 * half + 0];
            bm.y = xsum[n * XS_ + k0 + 2 * half + 1];
            c = __builtin_amdgcn_wmma_f32_16x16x4_f32(
                    false, a, false, bm, (short)0, c, false, false);
        }
        if (lane < 16)
            scores[(size_t)bb * NF_ + n0 + lane] = c[0] * (1.0f / FM_) + bias[0];
    }
}

// ------------------------------ CSR build ----------------------------------
// Deterministic inversion of segment_ids (shared across batch): thread c owns
// coarse node c, scans all NF nodes in order -> sorted member lists.
__global__ __launch_bounds__(256)
void k_csr(const int* __restrict__ seg, int* __restrict__ offsets,
           int* __restrict__ list)
{
    __shared__ int segL[NF_];
    __shared__ int cnt[NC_];
    __shared__ int ofs[NC_ + 1];
    const int t = threadIdx.x;
    #pragma unroll
    for (int i = 0; i < NF_ / 256; ++i) segL[i * 256 + t] = seg[i * 256 + t];
    __syncthreads();

    int k = 0;
    for (int n = 0; n < NF_; ++n) k += (segL[n] == t);
    cnt[t] = k;
    __syncthreads();

    if (t == 0) {
        int o = 0;
        for (int c = 0; c < NC_; ++c) { ofs[c] = o; o += cnt[c]; }
        ofs[NC_] = o;
    }
    __syncthreads();

    int pos = ofs[t];
    for (int n = 0; n < NF_; ++n)
        if (segL[n] == t) list[pos++] = n;

    offsets[t] = ofs[t];
    if (t == 0) offsets[NC_] = ofs[NC_];
}

// ------------------------------ Pass 2 -------------------------------------
__device__ __forceinline__ unsigned mono(float f) {
    unsigned u = __float_as_uint(f);
    return (u & 0x80000000u) ? ~u : (u | 0x80000000u);
}
__device__ __forceinline__ float unmono(unsigned u) {
    return (u & 0x80000000u) ? __uint_as_float(u & 0x7fffffffu)
                             : __uint_as_float(~u);
}

__global__ __launch_bounds__(256)
void k_weights(const float* __restrict__ scores, const int* __restrict__ seg,
               float* __restrict__ wbuf)
{
    __shared__ unsigned smax[NC_];
    __shared__ float    den[NC_];
    const int t  = threadIdx.x;
    const int bb = blockIdx.x;
    smax[t] = 0u;      // below mono(-inf); only read for segments with members
    den[t]  = 0.f;
    __syncthreads();

    float s[NF_ / 256]; float e[NF_ / 256]; int c[NF_ / 256];
    #pragma unroll
    for (int i = 0; i < NF_ / 256; ++i) {
        const int n = i * 256 + t;
        s[i] = scores[(size_t)bb * NF_ + n];
        c[i] = seg[n];
        atomicMax(&smax[c[i]], mono(s[i]));          // order-independent
    }
    __syncthreads();
    #pragma unroll
    for (int i = 0; i < NF_ / 256; ++i) {
        e[i] = __expf(s[i] - unmono(smax[c[i]]));
        atomicAdd(&den[c[i]], e[i]);
    }
    __syncthreads();
    #pragma unroll
    for (int i = 0; i < NF_ / 256; ++i) {
        const int n = i * 256 + t;
        wbuf[(size_t)bb * NF_ + n] = e[i] / den[c[i]];
    }
}

// ------------------------------ Pass 3 -------------------------------------
// One block per (coarse node, batch): register-accumulate the 4096-float tile
// over the sorted member list (deterministic order), prefetch next member.
__global__ __launch_bounds__(256)
void k_gather(const float* __restrict__ x, const float* __restrict__ wbuf,
              const int* __restrict__ offsets, const int* __restrict__ list,
              float* __restrict__ out)
{
    const int t  = threadIdx.x;
    const int c  = blockIdx.x;
    const int bb = blockIdx.y;
    const int beg = offsets[c], end = offsets[c + 1];

    float4 a0 = {0,0,0,0}, a1 = {0,0,0,0}, a2 = {0,0,0,0}, a3 = {0,0,0,0};

    for (int j = beg; j < end; ++j) {
        const int n = list[j];
        const float wn = wbuf[(size_t)bb * NF_ + n];
        const float4* base = (const float4*)(x + (size_t)(bb * NF_ + n) * TILE_);
        if (j + 1 < end) {           // global_prefetch_b8 the next 16KB tile
            const char* pf = (const char*)(x + (size_t)(bb * NF_ + list[j + 1]) * TILE_);
            __builtin_prefetch(pf + t * 64, 0, 1);
        }
        float4 v;
        v = base[0 * 256 + t];
        a0.x = fmaf(wn, v.x, a0.x); a0.y = fmaf(wn, v.y, a0.y);
        a0.z = fmaf(wn, v.z, a0.z); a0.w = fmaf(wn, v.w, a0.w);
        v = base[1 * 256 + t];
        a1.x = fmaf(wn, v.x, a1.x); a1.y = fmaf(wn, v.y, a1.y);
        a1.z = fmaf(wn, v.z, a1.z); a1.w = fmaf(wn, v.w, a1.w);
        v = base[2 * 256 + t];
        a2.x = fmaf(wn, v.x, a2.x); a2.y = fmaf(wn, v.y, a2.y);
        a2.z = fmaf(wn, v.z, a2.z); a2.w = fmaf(wn, v.w, a2.w);
        v = base[3 * 256 + t];
        a3.x = fmaf(wn, v.x, a3.x); a3.y = fmaf(wn, v.y, a3.y);
        a3.z = fmaf(wn, v.z, a3.z); a3.w = fmaf(wn, v.w, a3.w);
    }

    float4* ob = (float4*)(out + (size_t)(bb * NC_ + c) * TILE_);
    ob[0 * 256 + t] = a0;
    ob[1 * 256 + t] = a1;
    ob[2 * 256 + t] = a2;
    ob[3 * 256 + t] = a3;
}

// ------------------------------ Launch -------------------------------------
extern "C" void kernel_launch(void* const* d_in, const int* in_sizes, int n_in,
                              void* d_out, int out_size, void* d_ws, size_t ws_size,
                              hipStream_t stream)
{
    const float* x    = (const float*)d_in[0];   // [B, NF, Fm, H]
    const float* W    = (const float*)d_in[1];   // [1, H]
    const float* bias = (const float*)d_in[2];   // [1]
    const int*   seg  = (const int*)d_in[3];     // [NF]
    float*       out  = (float*)d_out;           // [B, NC, Fm, H]

    char* ws = (char*)d_ws;
    float* scores  = (float*)(ws);                       // B*NF floats  (64KB)
    float* wbuf    = (float*)(ws + 65536);               // B*NF floats  (64KB)
    int*   offsets = (int*)  (ws + 131072);              // NC+1 ints
    int*   list    = (int*)  (ws + 131072 + 2048);       // NF ints

    k_scores <<<dim3(NF_ / NPB_, B_), 256, 0, stream>>>(x, W, bias, scores);
    k_csr    <<<1, 256, 0, stream>>>(seg, offsets, list);
    k_weights<<<B_, 256, 0, stream>>>(scores, seg, wbuf);
    k_gather <<<dim3(NC_, B_), 256, 0, stream>>>(x, wbuf, offsets, list, out);
}